// ChamferDistanceLoss_2173253452340
// MI455X (gfx1250) — compile-verified
//
#include <hip/hip_runtime.h>
#include <math.h>

typedef float v2f __attribute__((ext_vector_type(2)));
typedef float v8f __attribute__((ext_vector_type(8)));

#define NPTS 16384
#define TILE 16
#define NTILES (NPTS / TILE)      // 1024 column tiles per batch
#define WPB 8                     // waves per block (256 threads)

// ---------------------------------------------------------------------------
// Pre-pack the WMMA B-operand payload for every point so the hot loop does a
// single b64 load per lane and zero VALU per tile.
//   pack[(b*2+0)*NPTS + n] = (-2x, -2y)         (lane half 0: K=0,1)
//   pack[(b*2+1)*NPTS + n] = (-2z, x^2+y^2+z^2) (lane half 1: K=2,3)
// ---------------------------------------------------------------------------
__global__ __launch_bounds__(256)
void pack_pts(const float* __restrict__ X, float* __restrict__ pack)
{
    const int i = blockIdx.x * 256 + threadIdx.x;   // point id in [0, 2*NPTS)
    if (i >= 2 * NPTS) return;
    const int b = i / NPTS, n = i % NPTS;
    const float* p = X + (size_t)i * 3;
    const float x = p[0], y = p[1], z = p[2];
    float* h0 = pack + ((size_t)(b * 2 + 0) * NPTS + n) * 2;
    float* h1 = pack + ((size_t)(b * 2 + 1) * NPTS + n) * 2;
    h0[0] = -2.0f * x;
    h0[1] = -2.0f * y;
    h1[0] = -2.0f * z;
    h1[1] = x * x + y * y + z * z;
}

// ---------------------------------------------------------------------------
// One wave owns a 16-row tile, streams all 16384 columns through
// V_WMMA_F32_16X16X4_F32, keeping a running elementwise min of squared
// distances in the accumulator layout.
//   D[m][n] = ||x_m||^2 - 2 x_m . y_n + ||y_n||^2 via
//   A[m]=[x0,x1,x2,1], B[:,n]=[-2y0,-2y1,-2y2,||y||^2] (pre-packed), C=||x_m||^2
// ---------------------------------------------------------------------------
__global__ __launch_bounds__(256)
void chamfer_nn_pass(const float* __restrict__ Prow,   // raw points owning rows
                     const float* __restrict__ Qpack,  // pre-packed column operands
                     float* __restrict__ out_min)      // B*NPTS squared-NN dists
{
    const int lane = threadIdx.x & 31;
    const int widx = threadIdx.x >> 5;
    const int gw   = blockIdx.x * WPB + widx;   // global wave id = row tile id
    const int b    = gw / NTILES;
    const int tile = gw % NTILES;
    const int row_base = tile * TILE;
    const int half = lane >> 4;                 // 0: K={0,1}; 1: K={2,3}
    const int lm   = lane & 15;                 // M (for A) / N (for B) in tile

    const float* Pb = Prow + (size_t)b * NPTS * 3;

    // ---- A operand (loop invariant): lane holds row m = row_base + lm ----
    const float* P = Pb + (size_t)(row_base + lm) * 3;
    const float px = P[0], py = P[1], pz = P[2];
    v2f a;
    a.x = half ? pz   : px;    // K = 2 : K = 0
    a.y = half ? 1.0f : py;    // K = 3 : K = 1

    // ---- C operand (loop invariant): c[v] = ||x_m||^2, m = row_base+8*half+v
    v8f cinit;
#pragma unroll
    for (int v = 0; v < 8; ++v) {
        const float* Pm = Pb + (size_t)(row_base + half * 8 + v) * 3;
        const float x = Pm[0], y = Pm[1], z = Pm[2];
        cinit[v] = x * x + y * y + z * z;
    }

    v8f vmin;
#pragma unroll
    for (int v = 0; v < 8; ++v) vmin[v] = 3.0e38f;

    // ---- stream all column tiles: 1 b64 load + 1 WMMA + 8 mins per tile ----
    const v2f* qp = (const v2f*)Qpack + (size_t)(b * 2 + half) * NPTS + lm;
#pragma unroll 8
    for (int ct = 0; ct < NTILES; ++ct) {
        const v2f bo = *qp;                     // (K0,K1) or (K2,K3) payload
        v8f d = __builtin_amdgcn_wmma_f32_16x16x4_f32(
            /*neg_a=*/false, a, /*neg_b=*/false, bo,
            /*c_mod=*/(short)0, cinit, /*reuse_a=*/false, /*reuse_b=*/false);
#pragma unroll
        for (int v = 0; v < 8; ++v) vmin[v] = fminf(vmin[v], d[v]);
        qp += TILE;
    }

    // ---- row-min: butterfly reduce across each 16-lane half, then store ----
#pragma unroll
    for (int v = 0; v < 8; ++v) {
        float m = vmin[v];
#pragma unroll
        for (int off = 1; off < 16; off <<= 1)
            m = fminf(m, __shfl_xor(m, off, 32));
        if (lm == 0)
            out_min[(size_t)b * NPTS + row_base + half * 8 + v] = m;
    }
}

// Final: sqrt(max(min_sq, EPS)), truncate at 0.5, masked means, sum.
__global__ __launch_bounds__(256)
void chamfer_reduce(const float* __restrict__ nn_sq, float* __restrict__ out)
{
    const int tid = threadIdx.x;
    float ssum = 0.f, scnt = 0.f, tsum = 0.f, tcnt = 0.f;
    for (int i = tid; i < 2 * NPTS; i += 256) {
        const float d = sqrtf(fmaxf(nn_sq[i], 1e-12f));
        if (d < 0.5f) { ssum += d; scnt += 1.0f; }
    }
    for (int i = tid; i < 2 * NPTS; i += 256) {
        const float d = sqrtf(fmaxf(nn_sq[2 * NPTS + i], 1e-12f));
        if (d < 0.5f) { tsum += d; tcnt += 1.0f; }
    }
    __shared__ float sm[4][256];
    sm[0][tid] = ssum; sm[1][tid] = scnt; sm[2][tid] = tsum; sm[3][tid] = tcnt;
    __syncthreads();
    for (int s = 128; s > 0; s >>= 1) {
        if (tid < s) {
#pragma unroll
            for (int k = 0; k < 4; ++k) sm[k][tid] += sm[k][tid + s];
        }
        __syncthreads();
    }
    if (tid == 0)
        out[0] = sm[0][0] / fmaxf(sm[1][0], 1.0f) +
                 sm[2][0] / fmaxf(sm[3][0], 1.0f);
}

extern "C" void kernel_launch(void* const* d_in, const int* in_sizes, int n_in,
                              void* d_out, int out_size, void* d_ws, size_t ws_size,
                              hipStream_t stream) {
    const float* src = (const float*)d_in[0];   // (2, 16384, 3) f32
    const float* tgt = (const float*)d_in[1];   // (2, 16384, 3) f32
    float* ws       = (float*)d_ws;
    float* src_nn   = ws;                        // 2*NPTS floats
    float* tgt_nn   = ws + 2 * NPTS;             // 2*NPTS floats
    float* tgt_pack = ws + 4 * NPTS;             // 2*2*NPTS*2 floats
    float* src_pack = ws + 4 * NPTS + 8 * NPTS;  // 2*2*NPTS*2 floats

    const dim3 block(256);
    const dim3 pgrid((2 * NPTS + 255) / 256);    // 128 blocks
    const dim3 grid(2 * NTILES / WPB);           // 2048 waves -> 256 blocks

    pack_pts<<<pgrid, block, 0, stream>>>(tgt, tgt_pack);
    pack_pts<<<pgrid, block, 0, stream>>>(src, src_pack);

    // Pass 1: waves own src rows vs packed tgt.  Pass 2: roles swapped.
    chamfer_nn_pass<<<grid, block, 0, stream>>>(src, tgt_pack, src_nn);
    chamfer_nn_pass<<<grid, block, 0, stream>>>(tgt, src_pack, tgt_nn);
    chamfer_reduce<<<1, block, 0, stream>>>(ws, (float*)d_out);
}